// FourierFuncTRF_47382079209936
// MI455X (gfx1250) — compile-verified
//
#include <hip/hip_runtime.h>

// ---------------------------------------------------------------------------
// Fourier basis transform as batched WMMA GEMM on gfx1250.
//   out[s,i,l,o] = sum_{b=0..63} basis[s,i,b,l] * coefs[i,o,1+b]
//                + const0 * coefs[i,o,0]
// basis[2(n-1)+0] = sin(2*pi*t*n/T)*scale, basis[2(n-1)+1] = cos(...)*scale
//
// Per i: GEMM  M=(S*L)=16384, K=64, N=64, f32 throughout via
// V_WMMA_F32_16X16X4_F32 (16 chained K=4 steps). Output is 134 MB streaming
// -> memory-bound at ~6 us on 23.3 TB/s HBM; f32 WMMA keeps us at that roof
// while preserving full reference precision (no f16 downcast needed).
// ---------------------------------------------------------------------------

typedef __attribute__((ext_vector_type(2))) float v2f;
typedef __attribute__((ext_vector_type(8))) float v8f;

#define NLAG   128
#define NBASIS 65
#define NOUT   64
#define NI     32
#define NS     128

__launch_bounds__(256)
__global__ void fourier_wmma_kernel(const float* __restrict__ x,
                                    const float* __restrict__ coefs,
                                    float* __restrict__ out) {
    // coefs[i] staged once per block: 64*65 f32 = 16640 B.
    // Row stride of 65 dwords is coprime with 64 banks -> conflict-free.
    __shared__ float sc[NOUT * NBASIS];

    const int bid = blockIdx.x;
    const int s   = bid >> 5;   // / NI
    const int i   = bid & 31;   // % NI
    const int tid = threadIdx.x;

    const float* csrc = coefs + (size_t)i * (NOUT * NBASIS);
    for (int idx = tid; idx < NOUT * NBASIS; idx += 256)
        sc[idx] = csrc[idx];
    __syncthreads();

    const int wave = tid >> 5;       // 8 waves: one 16-row l-tile each
    const int lane = tid & 31;
    const int hi   = lane >> 4;      // lane group: 0 -> K={0,1}, 1 -> K={2,3}
    const int col  = lane & 15;
    const int l0   = wave * 16;

    const float scale  = __builtin_sqrtf(2.0f / 127.0f);   // 1/sqrt(T/2)
    const float const0 = __builtin_sqrtf(1.0f / 127.0f);   // scale/sqrt(2)

    // t for this lane's A-matrix row (row index == col lane id)
    const float t = x[((size_t)(s * NI + i)) * NLAG + l0 + col];
    // v_sin/v_cos take revolutions: sin(2*pi*u) -> u = t*n/T
    const float u = t * (1.0f / 127.0f);

    // Build the 16x64 f32 A-tile in registers.
    // WMMA f32 16x16x4 A layout: lanes 0-15 hold (K=0 -> .x, K=1 -> .y),
    // lanes 16-31 hold (K=2 -> .x, K=3 -> .y). Basis index b = 4k + c with
    // b = 2(n-1) + {0:sin,1:cos}  =>  this lane's harmonic n = 2k + 1 + hi,
    // .x = sin, .y = cos. Only 16 sin + 16 cos per lane.
    v2f A[16];
#pragma unroll
    for (int k = 0; k < 16; ++k) {
        const float nf = (float)(2 * k + 1 + hi);
        const float ph = u * nf;
        A[k].x = __builtin_amdgcn_sinf(ph) * scale;
        A[k].y = __builtin_amdgcn_cosf(ph) * scale;
    }

    float* outbase = out + ((size_t)(s * NI + i) * NLAG + l0) * NOUT;

#pragma unroll
    for (int nb = 0; nb < 4; ++nb) {
        const int o = nb * 16 + col;

        // DC bias depends only on the output column -> identical in all 8
        // C VGPRs of this lane.
        const float bias = const0 * sc[o * NBASIS];
        v8f acc;
#pragma unroll
        for (int j = 0; j < 8; ++j) acc[j] = bias;

        // B operand 4x16 f32: lanes 0-15 rows K={0,1}, lanes 16-31 K={2,3};
        // B[k_abs, col] = coefs[i, o, 1 + k_abs] -> adjacent dwords in LDS.
        const float* bcol = &sc[o * NBASIS + 1 + 2 * hi];
#pragma unroll
        for (int k = 0; k < 16; ++k) {
            v2f b;
            b.x = bcol[4 * k + 0];
            b.y = bcol[4 * k + 1];
            acc = __builtin_amdgcn_wmma_f32_16x16x4_f32(
                /*neg_a=*/false, A[k], /*neg_b=*/false, b,
                /*c_mod=*/(short)0, acc, /*reuse_a=*/false, /*reuse_b=*/false);
        }

        // D layout: VGPR r, lanes 0-15 -> row r, lanes 16-31 -> row r+8.
        // 134 MB write-only stream -> non-temporal stores.
#pragma unroll
        for (int r = 0; r < 8; ++r) {
            const int rowM = r + hi * 8;
            __builtin_nontemporal_store(
                acc[r], &outbase[(size_t)rowM * NOUT + nb * 16 + col]);
        }
    }
}

extern "C" void kernel_launch(void* const* d_in, const int* in_sizes, int n_in,
                              void* d_out, int out_size, void* d_ws, size_t ws_size,
                              hipStream_t stream) {
    const float* x     = (const float*)d_in[0];
    const float* coefs = (const float*)d_in[1];
    float* out         = (float*)d_out;

    dim3 grid(NS * NI);   // one block per (s, i) pair: 4096 blocks
    dim3 block(256);      // 8 waves; each wave owns one 16-row l-tile
    hipLaunchKernelGGL(fourier_wmma_kernel, grid, block, 0, stream,
                       x, coefs, out);
}